// HeteroGNNModel_27015344292443
// MI455X (gfx1250) — compile-verified
//
#include <hip/hip_runtime.h>

// ---------------------------------------------------------------------------
// GraphSAGE 2-layer forward for MI455X (gfx1250, wave32).
//   out = sage2( relu( sage1(x) ) )
//   sage(x) = mean_agg(x) @ W_l + x @ W_r + b
//
// Bottleneck analysis: 102M f32 L2 atomics dominate; GEMMs (~0.6 GFLOP) run on
// the fp32 WMMA pipe (v_wmma_f32_16x16x4_f32) at full precision.
// Workspace layout (floats): agg[N*32] | deg[N] | h[N*32]  (~26 MB).
// ---------------------------------------------------------------------------

typedef __attribute__((ext_vector_type(2))) float v2f;
typedef __attribute__((ext_vector_type(8))) float v8f;

#define N_NODES 100000
#define F_DIM   32   // F_IN == F_HID == 32

// ------------------------------- zero fill ---------------------------------
__global__ void zero_f32(float* __restrict__ p, int n) {
  int i = blockIdx.x * blockDim.x + threadIdx.x;
  if (i < n) p[i] = 0.0f;
}

// --------------------------- edge scatter (mean num) -----------------------
// 8 threads per edge; each handles one float4 chunk of the 32-wide feature.
// Gathers are 128B-coalesced per edge; atomics land on 4 consecutive floats
// of one agg row -> good L2 atomic locality.
__global__ void sage_scatter(const int* __restrict__ src,
                             const int* __restrict__ dst,
                             const float* __restrict__ feat,  // [N][32]
                             float* __restrict__ agg,         // [N][32]
                             float* __restrict__ deg,         // [N]
                             int nEdges, int addDeg) {
  int t = blockIdx.x * blockDim.x + threadIdx.x;
  int work = nEdges * 8;
  if (t >= work) return;
  int e = t >> 3;
  int c = t & 7;
  int s = src[e];
  int d = dst[e];
  const float4 v = *(const float4*)(feat + (size_t)s * F_DIM + c * 4);
  float* a = agg + (size_t)d * F_DIM + c * 4;
  atomicAdd(a + 0, v.x);
  atomicAdd(a + 1, v.y);
  atomicAdd(a + 2, v.z);
  atomicAdd(a + 3, v.w);
  if (addDeg && c == 0) atomicAdd(deg + d, 1.0f);
}

// ------------------------------ WMMA GEMM ----------------------------------
// out[r, :] = act( (agg[r,:]*invdeg) @ Wl + feat[r,:] @ Wr + bias )
// One wave per 16-row tile. K=32 as 8 steps of v_wmma_f32_16x16x4_f32,
// two chained accumulations (mean*Wl then feat*Wr) per N-tile.
//
// Fragment layouts (ISA 7.12.2):
//   A 16x4 f32 : lane%16 = M, vgpr j holds K = j + 2*(lane>=16)
//   B  4x16 f32: lane%16 = N, vgpr j holds K = j + 2*(lane>=16)
//   D 16x16 f32: vgpr v holds M = v + 8*(lane>=16), N = lane%16
template <int OUT_W, bool RELU>
__global__ void sage_gemm(const float* __restrict__ feat,  // [N][32]
                          const float* __restrict__ agg,   // [N][32]
                          const float* __restrict__ deg,   // [N]
                          const float* __restrict__ Wl,    // [32][OUT_W]
                          const float* __restrict__ Wr,    // [32][OUT_W]
                          const float* __restrict__ bias,  // [OUT_W]
                          float* __restrict__ out,         // [N][OUT_W]
                          int nTiles) {
  int wave = (blockIdx.x * blockDim.x + threadIdx.x) >> 5;
  if (wave >= nTiles) return;  // wave-uniform: EXEC stays all-ones for WMMA

  int lane = threadIdx.x & 31;
  int half = lane >> 4;   // 0: K pair {0,1} / M rows 0..7 ; 1: {2,3} / 8..15
  int m    = lane & 15;
  int r0   = wave * 16;
  int row  = r0 + m;      // A-fragment row for this lane

  float invd = 1.0f / fmaxf(deg[row], 1.0f);
  const float* arow = agg  + (size_t)row * F_DIM;
  const float* xrow = feat + (size_t)row * F_DIM;

#pragma unroll
  for (int nt = 0; nt < OUT_W / 16; ++nt) {
    int col = nt * 16 + m;
    float bv = bias[col];
    v8f c = {bv, bv, bv, bv, bv, bv, bv, bv};

#pragma unroll
    for (int k0 = 0; k0 < 8; ++k0) {
      int K = k0 * 4 + half * 2;
      v2f aM, aX, bL, bR;
      aM.x = arow[K + 0] * invd;
      aM.y = arow[K + 1] * invd;
      aX.x = xrow[K + 0];
      aX.y = xrow[K + 1];
      bL.x = Wl[(K + 0) * OUT_W + col];
      bL.y = Wl[(K + 1) * OUT_W + col];
      bR.x = Wr[(K + 0) * OUT_W + col];
      bR.y = Wr[(K + 1) * OUT_W + col];
      // D = A*B + C ; accumulate mean@Wl and feat@Wr into the same C
      c = __builtin_amdgcn_wmma_f32_16x16x4_f32(
          /*neg_a=*/false, aM, /*neg_b=*/false, bL,
          /*c_mod=*/(short)0, c, /*reuse_a=*/false, /*reuse_b=*/false);
      c = __builtin_amdgcn_wmma_f32_16x16x4_f32(
          false, aX, false, bR, (short)0, c, false, false);
    }

#pragma unroll
    for (int v = 0; v < 8; ++v) {
      float o = c[v];
      if (RELU) o = fmaxf(o, 0.0f);
      out[(size_t)(r0 + v + 8 * half) * OUT_W + col] = o;
    }
  }
}

// ------------------------------- launcher ----------------------------------
extern "C" void kernel_launch(void* const* d_in, const int* in_sizes, int n_in,
                              void* d_out, int out_size, void* d_ws, size_t ws_size,
                              hipStream_t stream) {
  const float* x   = (const float*)d_in[0];
  const int*   ei  = (const int*)d_in[1];   // int32 (JAX x64 disabled)
  const float* W1l = (const float*)d_in[2];
  const float* W1r = (const float*)d_in[3];
  const float* b1  = (const float*)d_in[4];
  const float* W2l = (const float*)d_in[5];
  const float* W2r = (const float*)d_in[6];
  const float* b2  = (const float*)d_in[7];
  float* out = (float*)d_out;

  const int N = N_NODES;
  const int E = in_sizes[1] / 2;
  const int* src = ei;
  const int* dst = ei + E;

  float* agg = (float*)d_ws;        // N*32
  float* deg = agg + (size_t)N * F_DIM;  // N
  float* h   = deg + N;             // N*32

  const int TPB = 256;
  int nTiles = (N + 15) / 16;                        // 6250
  int gemmBlocks = (nTiles * 32 + TPB - 1) / TPB;    // 8 waves / block
  int scatBlocks = (E * 8 + TPB - 1) / TPB;

  // ---- layer 1: agg = sum x[src] -> dst, deg counts ----
  {
    int n = N * (F_DIM + 1);  // agg + deg contiguous
    zero_f32<<<(n + TPB - 1) / TPB, TPB, 0, stream>>>(agg, n);
  }
  sage_scatter<<<scatBlocks, TPB, 0, stream>>>(src, dst, x, agg, deg, E, 1);
  sage_gemm<32, true><<<gemmBlocks, TPB, 0, stream>>>(x, agg, deg, W1l, W1r, b1,
                                                      h, nTiles);

  // ---- layer 2: reuse agg (re-zero), deg unchanged (same graph) ----
  {
    int n = N * F_DIM;
    zero_f32<<<(n + TPB - 1) / TPB, TPB, 0, stream>>>(agg, n);
  }
  sage_scatter<<<scatBlocks, TPB, 0, stream>>>(src, dst, h, agg, deg, E, 0);
  sage_gemm<16, false><<<gemmBlocks, TPB, 0, stream>>>(h, agg, deg, W2l, W2r, b2,
                                                       out, nTiles);
}